// EndtoEndIntervetionMap_40003325395183
// MI455X (gfx1250) — compile-verified
//
#include <hip/hip_runtime.h>
#include <hip/hip_bf16.h>

typedef __attribute__((ext_vector_type(2))) float v2f;
typedef __attribute__((ext_vector_type(8))) float v8f;

#define SIR_GAMMA 0.1f
#define SIR_DT    0.5f      // 1.0 / STEPS, STEPS = 2
#define SIR_NITER 14        // WINDOW * STEPS = 7 * 2

// ReLU as exactly one v_max_num_f32 (IEEE maximumNumber; quiets sNaN per ISA),
// bypassing the backend's canonicalize (self-max) it inserts for fmaxf on
// values whose canonicality it can't prove (e.g. WMMA intrinsic results).
__device__ __forceinline__ float relu1(float x) {
    float r;
    asm("v_max_num_f32_e32 %0, 0, %1" : "=v"(r) : "v"(x));
    return r;
}

__global__ __launch_bounds__(256)
void sir_npi_wmma_kernel(const float* __restrict__ y0,
                         const float* __restrict__ x,
                         const float* __restrict__ W1,
                         const float* __restrict__ b1,
                         const float* __restrict__ W2,
                         const float* __restrict__ b2,
                         float* __restrict__ yout,
                         int numTiles /* tiles of 32 batch rows */) {
    const int lane   = threadIdx.x & 31;
    const int wvBlk  = threadIdx.x >> 5;
    const int wvsPB  = blockDim.x >> 5;
    const int gWave  = blockIdx.x * wvsPB + wvBlk;
    const int nWaves = gridDim.x * wvsPB;

    const int  hi  = (lane >= 16) ? 1 : 0;   // which 16-lane half
    const int  l15 = lane & 15;
    const bool b3  = (lane & 8) != 0;
    const bool b2l = (lane & 4) != 0;
    const bool b1l = (lane & 2) != 0;

    // ---- Preload W1 B-fragments (4 N-tiles x 2 K-steps), b1, W2, b2 ----
    // B 4x16 f32 layout (mirrors A 16x4): VGPR0 lanes0-15 = K row 0, lanes16-31
    // = K row 2; VGPR1 = K rows 1 / 3. W1 is row-major (8,64): [k][n] = k*64+n.
    v2f   Bfrag[4][2];
    float b1v[4], w2v[4];
#pragma unroll
    for (int t = 0; t < 4; ++t) {
        const int n = 16 * t + l15;
#pragma unroll
        for (int k = 0; k < 2; ++k) {
            const int kr = 4 * k + (hi ? 2 : 0);
            v2f bf;
            bf.x = W1[(kr + 0) * 64 + n];
            bf.y = W1[(kr + 1) * 64 + n];
            Bfrag[t][k] = bf;
        }
        b1v[t] = b1[n];
        w2v[t] = W2[n];
    }
    const float b2v = b2[0];

    for (int tile = gWave; tile < numTiles; tile += nWaves) {
        const int rowBase = tile * 32;
        float betaSub[2];   // one fully-reduced sigmoid per lane per sub-tile

#pragma unroll
        for (int sub = 0; sub < 2; ++sub) {
            const int rb = rowBase + 16 * sub;

            // ---- Layer 1: h[16x64] = x[16x8] @ W1[8x64] + b1, 8 WMMA f32 ----
            // Bias folded into C init: C column == l15, so splat b1v[t].
            v8f acc[4];
#pragma unroll
            for (int t = 0; t < 4; ++t) {
#pragma unroll
                for (int m = 0; m < 8; ++m) acc[t][m] = b1v[t];
            }

#pragma unroll
            for (int k = 0; k < 2; ++k) {
                // A 16x4 fragment: lanes0-15 K={4k,4k+1}, lanes16-31 K={4k+2,4k+3}
                const float* xr = x + (size_t)(rb + l15) * 8 + 4 * k + 2 * hi;
                const v2f a = *(const v2f*)xr;   // 8B-aligned
#pragma unroll
                for (int t = 0; t < 4; ++t) {
                    acc[t] = __builtin_amdgcn_wmma_f32_16x16x4_f32(
                        /*neg_a=*/false, a, /*neg_b=*/false, Bfrag[t][k],
                        /*c_mod=*/(short)0, acc[t],
                        /*reuse_a=*/false, /*reuse_b=*/false);
                }
            }

            // ---- relu (1x v_max_num) + layer-2 partial dot ----
            // C layout: VGPR m, lanes0-15 = row m, lanes16-31 = row m+8, N = l15.
            float part[8];
#pragma unroll
            for (int m = 0; m < 8; ++m) part[m] = 0.0f;
#pragma unroll
            for (int t = 0; t < 4; ++t) {
#pragma unroll
                for (int m = 0; m < 8; ++m) {
                    part[m] = fmaf(relu1(acc[t][m]), w2v[t], part[m]);
                }
            }

            // ---- reduce-SCATTER over the 16-lane half (masks 8,4,2 select,
            //      mask 1 plain). Ends with one row-sum per lane:
            //      row_in_sub = 8*hi + 4*b3 + 2*b2l + b1l ----
            float k4[4];
#pragma unroll
            for (int m = 0; m < 4; ++m) {
                const float send = b3 ? part[m] : part[m + 4];
                const float keep = b3 ? part[m + 4] : part[m];
                k4[m] = keep + __shfl_xor(send, 8, 32);
            }
            float k2[2];
#pragma unroll
            for (int m = 0; m < 2; ++m) {
                const float send = b2l ? k4[m] : k4[m + 2];
                const float keep = b2l ? k4[m + 2] : k4[m];
                k2[m] = keep + __shfl_xor(send, 4, 32);
            }
            {
                const float send = b1l ? k2[0] : k2[1];
                const float keep = b1l ? k2[1] : k2[0];
                float k1 = keep + __shfl_xor(send, 2, 32);
                k1 += __shfl_xor(k1, 1, 32);
                const float z = k1 + b2v;
                betaSub[sub] = 1.0f / (1.0f + __expf(-z));  // sigmoid
            }
        }

        // ---- lane -> row mapping after reduce-scatter ----
        // Lane holds rows {rloc, 16+rloc} where rloc = (lane>>1)&15;
        // lane bit0 picks which sub-tile's row this lane integrates.
        const int   rloc = (lane >> 1) & 15;
        const int   row  = rowBase + ((lane & 1) ? 16 : 0) + rloc;
        const float beta = (lane & 1) ? betaSub[1] : betaSub[0];

        // ---- SIR explicit Euler, 14 steps, one trajectory per lane.
        // dS = -bd*S*I ; dI = bd*S*I - (dt*gamma)*I ; R by conservation. ----
        const size_t r = (size_t)row * 3;
        const float S0 = y0[r + 0];
        const float I0 = y0[r + 1];
        const float R0 = y0[r + 2];
        const float total = S0 + I0 + R0;
        const float bd = beta * SIR_DT;                  // dt*beta
        const float im = 1.0f - SIR_DT * SIR_GAMMA;      // 0.95
        float S = S0, I = I0;
#pragma unroll
        for (int it = 0; it < SIR_NITER; ++it) {
            const float inf = bd * (S * I);              // dt*beta*S*I
            S = S - inf;
            I = fmaf(I, im, inf);                        // I*(1-dt*gamma)+inf
        }
        yout[r + 0] = S;
        yout[r + 1] = I;
        yout[r + 2] = (total - S) - I;                   // conservation
    }
}

extern "C" void kernel_launch(void* const* d_in, const int* in_sizes, int n_in,
                              void* d_out, int out_size, void* d_ws, size_t ws_size,
                              hipStream_t stream) {
    const float* y0 = (const float*)d_in[0];  // (B, 3)
    const float* x  = (const float*)d_in[1];  // (B, 8)
    const float* W1 = (const float*)d_in[2];  // (8, 64)
    const float* b1 = (const float*)d_in[3];  // (64,)
    const float* W2 = (const float*)d_in[4];  // (64, 1)
    const float* b2 = (const float*)d_in[5];  // (1,)
    float* yout = (float*)d_out;              // (B, 3)

    const int B = in_sizes[0] / 3;            // 4194304; 32 | B
    const int numTiles = B / 32;              // 131072 wave-tiles

    const int threads = 256;                  // 8 waves per block
    const int wavesPerBlock = threads / 32;
    int blocks = (numTiles + wavesPerBlock - 1) / wavesPerBlock;
    if (blocks > 8192) blocks = 8192;         // grid-stride over tiles

    sir_npi_wmma_kernel<<<blocks, threads, 0, stream>>>(
        y0, x, W1, b1, W2, b2, yout, numTiles);
}